// SimpleGCNLayer_43946105372999
// MI455X (gfx1250) — compile-verified
//
#include <hip/hip_runtime.h>
#include <hip/hip_bf16.h>

// ---------------------------------------------------------------------------
// SimpleGCNLayer on MI455X (gfx1250, wave32)
//
// Phases:
//  1) h = x @ W^T + b        -> fp32 WMMA (v_wmma_f32_16x16x4_f32), h in ws
//  2) deg[dst] += w          -> global_atomic_add_f32
//  3) dinv = min(rsqrt(deg), 1e4)
//  4) agg[dst] += h[src]*norm -> half-wave per edge, float4 gather, f32 atomics
//     (h is 25.6 MB -> resident in 192 MB L2; atomics resolve in L2)
//  5) relu in place on d_out
// ---------------------------------------------------------------------------

typedef float v2f __attribute__((ext_vector_type(2)));
typedef float v8f __attribute__((ext_vector_type(8)));

#define D_FEAT 64   // D_IN == D_OUT == 64 in the reference

// ---------------- zero fill ----------------
__global__ void gcn_zero_kernel(float* __restrict__ p, long n) {
    long i = (long)blockIdx.x * blockDim.x + threadIdx.x;
    long stride = (long)gridDim.x * blockDim.x;
    for (; i < n; i += stride) p[i] = 0.0f;
}

// ---------------- h = x @ W^T + b via fp32 WMMA ----------------
// One wave computes one 16x16 tile of h. K accumulated 4 at a time with
// V_WMMA_F32_16X16X4_F32 (exact fp32 -> matches reference precision).
//
// fp32 A (16x4, MxK) layout: lane L<16 holds (M=L, K=0..1) in v[0..1];
//                            lane L>=16 holds (M=L-16, K=2..3).
// fp32 B (4x16, KxN) layout (row striped across lanes per VGPR):
//   v0: lanes 0-15 = row K=0, lanes 16-31 = row K=2; v1: K=1 / K=3.
// => both fragments are a single float2 load per lane per k-step.
// C/D (16x16 f32): VGPR v, lanes 0-15 -> (M=v, N=lane); lanes 16-31 -> (M=v+8).
__global__ void __launch_bounds__(256)
gcn_gemm_wmma_kernel(const float* __restrict__ x,   // [N, 64]
                     const float* __restrict__ W,   // [64, 64] row-major (out, in)
                     const float* __restrict__ b,   // [64]
                     float* __restrict__ h,         // [N, 64]
                     int n_tiles)                   // (N/16) * (64/16)
{
    int wave = (int)((blockIdx.x * blockDim.x + threadIdx.x) >> 5);
    if (wave >= n_tiles) return;           // wave-uniform guard (EXEC stays all-1s)

    int lane = threadIdx.x & 31;
    int half = lane >> 4;                  // 0: lanes 0-15, 1: lanes 16-31
    int q    = lane & 15;

    int m0 = (wave >> 2) * 16;             // 4 col-tiles per row-tile (64/16)
    int n0 = (wave & 3) * 16;

    // Per-lane base pointers; k-step advances by 4 floats.
    const float* arow = x + (long)(m0 + q) * D_FEAT + half * 2;   // x[m][k..k+1]
    const float* brow = W + (long)(n0 + q) * D_FEAT + half * 2;   // W[n][k..k+1] = B[k][n]

    v8f c = {};
#pragma unroll
    for (int k0 = 0; k0 < D_FEAT; k0 += 4) {
        v2f a  = *(const v2f*)(arow + k0);
        v2f bb = *(const v2f*)(brow + k0);
        // (neg_a, A, neg_b, B, c_mod, C, reuse_a, reuse_b)
        c = __builtin_amdgcn_wmma_f32_16x16x4_f32(
                false, a, false, bb, (short)0, c, false, false);
    }

    float bias = b[n0 + q];                // column bias, same for all 8 rows
    float* outp = h + (long)(m0 + half * 8) * D_FEAT + n0 + q;
#pragma unroll
    for (int v = 0; v < 8; ++v)
        outp[(long)v * D_FEAT] = c[v] + bias;
}

// ---------------- deg[dst] += edge_weight ----------------
__global__ void gcn_deg_kernel(const int* __restrict__ dst,
                               const float* __restrict__ w,
                               float* __restrict__ deg, int E) {
    int e = blockIdx.x * blockDim.x + threadIdx.x;
    if (e < E) atomicAdd(&deg[dst[e]], w[e]);
}

// ---------------- dinv = min(deg^-0.5, 10000) ----------------
__global__ void gcn_dinv_kernel(const float* __restrict__ deg,
                                float* __restrict__ dinv, int N) {
    int i = blockIdx.x * blockDim.x + threadIdx.x;
    if (i < N) {
        float r = rsqrtf(deg[i]);          // deg==0 -> +inf -> clamp
        dinv[i] = fminf(r, 10000.0f);
    }
}

// ---------------- agg[dst] += h[src] * norm ----------------
// 16 lanes (half wave) per edge, 4 features per lane (float4 gather).
// Sub-group leader (lane%16==0) loads edge meta + computes norm; __shfl
// broadcast within the wave32 (leaders at lanes 0 and 16).
__global__ void __launch_bounds__(256)
gcn_scatter_kernel(const int* __restrict__ src,
                   const int* __restrict__ dst,
                   const float* __restrict__ w,
                   const float* __restrict__ dinv,
                   const float* __restrict__ h,     // [N, 64]
                   float* __restrict__ agg,         // [N, 64] (d_out accumulator)
                   int E)
{
    int tid  = blockIdx.x * blockDim.x + threadIdx.x;
    int lane = threadIdx.x & 31;
    int leader = lane & 16;                 // 0 or 16: broadcast source lane
    long e = ((long)(tid >> 5) << 1) + (lane >> 4);   // 2 edges per wave
    int fl = (lane & 15) << 2;              // feature base: 0,4,...,60

    int s = 0, d = 0; float norm = 0.0f;
    if (((lane & 15) == 0) && e < (long)E) {
        s = src[e];
        d = dst[e];
        norm = dinv[s] * w[e] * dinv[d];
    }
    s    = __shfl(s, leader);
    d    = __shfl(d, leader);
    norm = __shfl(norm, leader);

    if (e < (long)E) {
        const float4 hv = *(const float4*)(h + (long)s * D_FEAT + fl);
        float* ap = agg + (long)d * D_FEAT + fl;
        atomicAdd(ap + 0, hv.x * norm);
        atomicAdd(ap + 1, hv.y * norm);
        atomicAdd(ap + 2, hv.z * norm);
        atomicAdd(ap + 3, hv.w * norm);
    }
}

// ---------------- relu in place ----------------
__global__ void gcn_relu_kernel(float* __restrict__ p, long n) {
    long i = (long)blockIdx.x * blockDim.x + threadIdx.x;
    long stride = (long)gridDim.x * blockDim.x;
    for (; i < n; i += stride) p[i] = fmaxf(p[i], 0.0f);
}

// ---------------------------------------------------------------------------
extern "C" void kernel_launch(void* const* d_in, const int* in_sizes, int n_in,
                              void* d_out, int out_size, void* d_ws, size_t ws_size,
                              hipStream_t stream) {
    // Inputs (setup_inputs order): x, edge_index, edge_weight, W, b
    const float* x   = (const float*)d_in[0];      // [N, 64] f32
    const int*   ei  = (const int*)d_in[1];        // [2, E]  (jax default int32)
    const float* w   = (const float*)d_in[2];      // [E]     f32
    const float* W   = (const float*)d_in[3];      // [64,64] f32
    const float* b   = (const float*)d_in[4];      // [64]    f32
    float* out = (float*)d_out;                    // [N, 64] f32

    const int N = in_sizes[0] / D_FEAT;            // 100000
    const int E = in_sizes[2];                     // 1600000
    const int* src = ei;                           // row 0 of edge_index
    const int* dst = ei + E;                       // row 1 of edge_index

    // Workspace layout: h [N*64 f32] | deg [N f32] | dinv [N f32]  (~26.4 MB)
    char*  ws   = (char*)d_ws;
    float* h    = (float*)ws;
    float* deg  = (float*)(ws + (size_t)N * D_FEAT * sizeof(float));
    float* dinv = deg + N;

    // 1) zero accumulators (harness poisons buffers; we must init every call)
    gcn_zero_kernel<<<2048, 256, 0, stream>>>(out, (long)N * D_FEAT);
    gcn_zero_kernel<<<512, 256, 0, stream>>>(deg, (long)N);

    // 2) h = x @ W^T + b  (fp32 WMMA).  N=100000 is a multiple of 16.
    const int n_tiles = (N / 16) * (D_FEAT / 16);          // 25000
    const int gemm_blocks = (n_tiles + 7) / 8;             // 8 waves / block
    gcn_gemm_wmma_kernel<<<gemm_blocks, 256, 0, stream>>>(x, W, b, h, n_tiles);

    // 3) degree + inverse-sqrt clamp
    gcn_deg_kernel<<<(E + 255) / 256, 256, 0, stream>>>(dst, w, deg, E);
    gcn_dinv_kernel<<<(N + 255) / 256, 256, 0, stream>>>(deg, dinv, N);

    // 4) edge gather/scale/scatter: 16 threads per edge
    long sthreads = (long)E * 16;
    gcn_scatter_kernel<<<(int)((sthreads + 255) / 256), 256, 0, stream>>>(
        src, dst, w, dinv, h, out, E);

    // 5) relu in place
    gcn_relu_kernel<<<2048, 256, 0, stream>>>(out, (long)N * D_FEAT);
}